// SheafGluingCG_76381698392307
// MI455X (gfx1250) — compile-verified
//
#include <hip/hip_runtime.h>
#include <hip/hip_bf16.h>
#include <math.h>

// ---------------- problem constants (match reference) ----------------
#define B_    2
#define M_    25000
#define D_    16
#define E_    400000
#define A_    8
#define FEDGE 16
#define HEDGE 32
#define MD_   (M_ * D_)      // 400000
#define BMD_  (B_ * MD_)     // 800000
#define NIT   20

typedef __attribute__((ext_vector_type(2))) float v2f;
typedef __attribute__((ext_vector_type(8))) float v8f;

// ---------------- small helpers ----------------
__device__ __forceinline__ float dot4(float4 a, float4 b) {
    return a.x * b.x + a.y * b.y + a.z * b.z + a.w * b.w;
}
__device__ __forceinline__ float4 fma4(float4 a, float s, float4 c) {
    c.x = fmaf(a.x, s, c.x); c.y = fmaf(a.y, s, c.y);
    c.z = fmaf(a.z, s, c.z); c.w = fmaf(a.w, s, c.w);
    return c;
}
__device__ __forceinline__ void gatomic_add(float* p, float v) {
    __hip_atomic_fetch_add(p, v, __ATOMIC_RELAXED, __HIP_MEMORY_SCOPE_AGENT);
}
__device__ __forceinline__ void at4(float* p, float4 v, float sgn) {
    gatomic_add(p + 0, sgn * v.x);
    gatomic_add(p + 1, sgn * v.y);
    gatomic_add(p + 2, sgn * v.z);
    gatomic_add(p + 3, sgn * v.w);
}
// block (256 threads) tree reduction, then one global atomic
__device__ __forceinline__ void block_reduce_atomic(float v, float* target) {
    __shared__ float s[256];
    int t = threadIdx.x;
    s[t] = v;
    __syncthreads();
    for (int off = 128; off > 0; off >>= 1) {
        if (t < off) s[t] += s[t + off];
        __syncthreads();
    }
    if (t == 0) gatomic_add(target, s[0]);
}

// ---------------- edge-weight MLP via V_WMMA_F32_16X16X4_F32 ----------------
// One wave (32 lanes) computes H = edge_feat[16 edges,16] @ W1[16,32] as two
// 16x16 f32 WMMA accumulator tiles (K=16 chained as 4x K=4), applies bias+silu,
// dots with W2[32] via xor-shuffle reduction, then softplus -> w[e].
__global__ __launch_bounds__(32)
void k_edge_w(const float* __restrict__ feat, const float* __restrict__ W1,
              const float* __restrict__ b1, const float* __restrict__ W2,
              const float* __restrict__ b2, float* __restrict__ wout) {
    int base = blockIdx.x * 16;
    if (base >= E_) return;                 // uniform per wave: EXEC stays all-ones
    int lane = threadIdx.x;
    int m    = lane & 15;                   // N column (B/C/D) and M row (A)
    int hi   = lane >> 4;                   // 0: K=k0,k0+1  1: K=k0+2,k0+3

    v8f acc0 = {0.f, 0.f, 0.f, 0.f, 0.f, 0.f, 0.f, 0.f};   // N = 0..15
    v8f acc1 = {0.f, 0.f, 0.f, 0.f, 0.f, 0.f, 0.f, 0.f};   // N = 16..31
#pragma unroll
    for (int k0 = 0; k0 < FEDGE; k0 += 4) {
        int klo = k0 + hi * 2;
        const float* fr = feat + (size_t)(base + m) * FEDGE + klo;
        v2f a; a.x = fr[0]; a.y = fr[1];                      // A: 16x4 tile
        v2f bA, bB;                                           // B: 4x16 tiles
        bA.x = W1[(size_t)klo * HEDGE + m];
        bA.y = W1[(size_t)(klo + 1) * HEDGE + m];
        bB.x = W1[(size_t)klo * HEDGE + 16 + m];
        bB.y = W1[(size_t)(klo + 1) * HEDGE + 16 + m];
        acc0 = __builtin_amdgcn_wmma_f32_16x16x4_f32(false, a, false, bA,
                                                     (short)0, acc0, false, false);
        acc1 = __builtin_amdgcn_wmma_f32_16x16x4_f32(false, a, false, bB,
                                                     (short)0, acc1, false, false);
    }

    float b1lo = b1[m], b1hi = b1[16 + m];
    float w2lo = W2[m], w2hi = W2[16 + m];
    float rowsum[8];
#pragma unroll
    for (int v = 0; v < 8; ++v) {
        float h0 = acc0[v] + b1lo;                   // silu
        float h1 = acc1[v] + b1hi;
        h0 = h0 / (1.f + expf(-h0));
        h1 = h1 / (1.f + expf(-h1));
        float t = h0 * w2lo + h1 * w2hi;             // partial of h @ W2
        t += __shfl_xor(t, 1, 32);
        t += __shfl_xor(t, 2, 32);
        t += __shfl_xor(t, 4, 32);
        t += __shfl_xor(t, 8, 32);                   // reduce within 16-lane half
        rowsum[v] = t;
    }
    if (m == 0) {
#pragma unroll
        for (int v = 0; v < 8; ++v) {
            int e = base + v + hi * 8;               // D tile row mapping
            float z  = rowsum[v] + b2[0];
            float sp = (z > 20.f) ? z : log1pf(expf(z));   // softplus
            wout[e]  = sp + 1e-6f;
        }
    }
}

// ---------------- edge scatter pass (the bandwidth-bound matvec core) -------
// 4 lanes per edge (lane sub = d-quad). R kept in VGPRs; r[a] reduced with two
// shfl_xor steps; scatter via global f32 atomics, 4 consecutive d per lane so
// the 4-lane group covers a contiguous 64B node row.
__global__ __launch_bounds__(256)
void k_edge_apply(const float* __restrict__ pin, float* __restrict__ out,
                  const float* __restrict__ Rsrc, const float* __restrict__ Rdst,
                  const float* __restrict__ wE, const int* __restrict__ srcIdx,
                  const int* __restrict__ dstIdx) {
    int tid = blockIdx.x * 256 + threadIdx.x;
    int e   = tid >> 2;
    if (e >= E_) return;
    int sub = tid & 3;

    int s = srcIdx[e];
    int t = dstIdx[e];
    float wgt = wE[e];

    const float* ps0p = pin + (size_t)s * D_ + sub * 4;
    const float* pd0p = pin + (size_t)t * D_ + sub * 4;
    float4 ps0 = *(const float4*)ps0p;
    float4 ps1 = *(const float4*)(ps0p + MD_);
    float4 pd0 = *(const float4*)pd0p;
    float4 pd1 = *(const float4*)(pd0p + MD_);

    const float* Rs = Rsrc + (size_t)e * (A_ * D_) + sub * 4;
    const float* Rd = Rdst + (size_t)e * (A_ * D_) + sub * 4;
    // stream-ahead prefetch of R (gfx1250 global_prefetch_b8)
    __builtin_prefetch(Rs + (size_t)64 * (A_ * D_), 0, 0);
    __builtin_prefetch(Rd + (size_t)64 * (A_ * D_), 0, 0);

    float4 rs[A_], rd[A_];
    float pr0[A_], pr1[A_];
#pragma unroll
    for (int a = 0; a < A_; ++a) {
        rs[a] = *(const float4*)(Rs + a * D_);
        rd[a] = *(const float4*)(Rd + a * D_);
        pr0[a] = dot4(rs[a], ps0) - dot4(rd[a], pd0);
        pr1[a] = dot4(rs[a], ps1) - dot4(rd[a], pd1);
    }
    // reduce partial dots across the 4-lane d-group; all 4 lanes end with full r[a]
#pragma unroll
    for (int a = 0; a < A_; ++a) {
        float v0 = pr0[a];
        v0 += __shfl_xor(v0, 1, 32);
        v0 += __shfl_xor(v0, 2, 32);
        pr0[a] = v0 * wgt;
        float v1 = pr1[a];
        v1 += __shfl_xor(v1, 1, 32);
        v1 += __shfl_xor(v1, 2, 32);
        pr1[a] = v1 * wgt;
    }
    float4 z = {0.f, 0.f, 0.f, 0.f};
    float4 cs0 = z, cs1 = z, cd0 = z, cd1 = z;
#pragma unroll
    for (int a = 0; a < A_; ++a) {
        cs0 = fma4(rs[a], pr0[a], cs0);
        cs1 = fma4(rs[a], pr1[a], cs1);
        cd0 = fma4(rd[a], pr0[a], cd0);
        cd1 = fma4(rd[a], pr1[a], cd1);
    }
    // acc[src] += R_src^T r ; acc[dst] += -R_dst^T r ; LAM = 1
    float* os = out + (size_t)s * D_ + sub * 4;
    float* od = out + (size_t)t * D_ + sub * 4;
    at4(os,        cs0, 1.f);
    at4(os + MD_,  cs1, 1.f);
    at4(od,        cd0, -1.f);
    at4(od + MD_,  cd1, -1.f);
}

// ---------------- CG elementwise / reduction kernels ----------------
__global__ __launch_bounds__(256)
void k_copy(float* __restrict__ d, const float* __restrict__ srcv, int n) {
    int i = blockIdx.x * 256 + threadIdx.x;
    if (i < n) d[i] = srcv[i];
}

__global__ __launch_bounds__(16)
void k_zero16(float* sc) { sc[threadIdx.x] = 0.f; }

// r = c0 - Ap ; p = r ; rsold[b] += r*r   (grid.y = batch)
__global__ __launch_bounds__(256)
void k_resid_init(const float* __restrict__ c0, const float* __restrict__ Ap,
                  float* __restrict__ r, float* __restrict__ p,
                  float* __restrict__ rsold) {
    int by = blockIdx.y;
    int i  = blockIdx.x * 256 + threadIdx.x;
    float v = 0.f;
    if (i < MD_) {
        size_t idx = (size_t)by * MD_ + i;
        float rr = c0[idx] - Ap[idx];
        r[idx] = rr;
        p[idx] = rr;
        v = rr * rr;
    }
    block_reduce_atomic(v, rsold + by);
}

// out[b] += a.b   (grid.y = batch)
__global__ __launch_bounds__(256)
void k_dot(const float* __restrict__ a, const float* __restrict__ b,
           float* __restrict__ outslot) {
    int by = blockIdx.y;
    int i  = blockIdx.x * 256 + threadIdx.x;
    float v = 0.f;
    if (i < MD_) {
        size_t idx = (size_t)by * MD_ + i;
        v = a[idx] * b[idx];
    }
    block_reduce_atomic(v, outslot + by);
}

// scalar slots: 0,1 rsold | 2,3 pAp | 4,5 alpha | 6,7 rsnew | 8,9 beta
__global__ __launch_bounds__(2)
void k_alpha(float* sc) {
    int b = threadIdx.x;
    sc[4 + b] = sc[0 + b] / (sc[2 + b] + 1e-12f);
    sc[6 + b] = 0.f;                       // zero rsnew for fused accumulation
}

// x += alpha*p ; r -= alpha*Ap ; rsnew[b] += r*r  (fused, grid.y = batch)
__global__ __launch_bounds__(256)
void k_update_xr(float* __restrict__ x, float* __restrict__ r,
                 const float* __restrict__ p, const float* __restrict__ Ap,
                 const float* __restrict__ sc) {
    int by = blockIdx.y;
    int i  = blockIdx.x * 256 + threadIdx.x;
    float alpha = sc[4 + by];
    float v = 0.f;
    if (i < MD_) {
        size_t idx = (size_t)by * MD_ + i;
        x[idx] = fmaf(alpha, p[idx], x[idx]);
        float rn = fmaf(-alpha, Ap[idx], r[idx]);
        r[idx] = rn;
        v = rn * rn;
    }
    block_reduce_atomic(v, (float*)sc + 6 + by);
}

__global__ __launch_bounds__(2)
void k_beta(float* sc) {
    int b = threadIdx.x;
    sc[8 + b] = sc[6 + b] / (sc[0 + b] + 1e-12f);
    sc[0 + b] = sc[6 + b];                 // rsold = rsnew
    sc[2 + b] = 0.f;                       // zero pAp for next iteration
}

// p = r + beta*p   (grid.y = batch)
__global__ __launch_bounds__(256)
void k_update_p(float* __restrict__ p, const float* __restrict__ r,
                const float* __restrict__ sc) {
    int by = blockIdx.y;
    int i  = blockIdx.x * 256 + threadIdx.x;
    if (i < MD_) {
        size_t idx = (size_t)by * MD_ + i;
        p[idx] = fmaf(sc[8 + by], p[idx], r[idx]);
    }
}

// ---------------- host-side fixed launch sequence (graph-capture safe) ------
extern "C" void kernel_launch(void* const* d_in, const int* in_sizes, int n_in,
                              void* d_out, int out_size, void* d_ws, size_t ws_size,
                              hipStream_t stream) {
    const float* c0   = (const float*)d_in[0];
    const float* Rsrc = (const float*)d_in[1];
    const float* Rdst = (const float*)d_in[2];
    const float* feat = (const float*)d_in[3];
    const float* W1   = (const float*)d_in[4];
    const float* b1   = (const float*)d_in[5];
    const float* W2   = (const float*)d_in[6];
    const float* b2   = (const float*)d_in[7];
    const int*   srcI = (const int*)d_in[8];
    const int*   dstI = (const int*)d_in[9];
    float* x = (float*)d_out;

    // workspace layout (all float4-aligned given 256B-aligned d_ws)
    float* sc = (float*)d_ws;          // 16 scalar slots
    float* w  = sc + 16;               // E edge weights
    float* p  = w + E_;                // B*M*D
    float* r  = p + BMD_;              // B*M*D
    float* Ap = r + BMD_;              // B*M*D

    const dim3 blk256(256);
    const dim3 gEdge(E_ * 4 / 256);                // 6250
    const dim3 gCopy((BMD_ + 255) / 256);          // 3125
    const dim3 gVec((MD_ + 255) / 256, B_);        // 1563 x 2

    // --- setup ---
    k_zero16<<<1, 16, 0, stream>>>(sc);
    k_edge_w<<<E_ / 16, 32, 0, stream>>>(feat, W1, b1, W2, b2, w);
    k_copy<<<gCopy, blk256, 0, stream>>>(x, c0, BMD_);       // x = c0
    k_copy<<<gCopy, blk256, 0, stream>>>(Ap, c0, BMD_);      // Ap = c0 (matvec base)
    k_edge_apply<<<gEdge, blk256, 0, stream>>>(c0, Ap, Rsrc, Rdst, w, srcI, dstI);
    k_resid_init<<<gVec, blk256, 0, stream>>>(c0, Ap, r, p, sc + 0);

    // --- 20 CG iterations ---
    for (int it = 0; it < NIT; ++it) {
        k_copy<<<gCopy, blk256, 0, stream>>>(Ap, p, BMD_);   // Ap = p
        k_edge_apply<<<gEdge, blk256, 0, stream>>>(p, Ap, Rsrc, Rdst, w, srcI, dstI);
        k_dot<<<gVec, blk256, 0, stream>>>(p, Ap, sc + 2);   // pAp
        k_alpha<<<1, 2, 0, stream>>>(sc);
        k_update_xr<<<gVec, blk256, 0, stream>>>(x, r, p, Ap, sc);
        k_beta<<<1, 2, 0, stream>>>(sc);
        k_update_p<<<gVec, blk256, 0, stream>>>(p, r, sc);
    }
}